// UniformAssigner_72353019068755
// MI455X (gfx1250) — compile-verified
//
#include <hip/hip_runtime.h>
#include <climits>

// ---------------------------------------------------------------------------
// UniformAssigner for MI455X (gfx1250, wave32).
//   in[0] grid_bboxes  [N,4] f32 xyxy   (N = 200000)
//   in[1] gt_bboxes    [M,4] f32 xyxy   (M = 128)
//   in[2] gt_labels    [M]   int
//   out: labels [N] (as f32) ++ bboxes [N,4] f32
// ---------------------------------------------------------------------------

#define UA_POS_THR 0.15f
#define UA_NEG_THR 0.70f
#define UA_TPB_TOPK 512
#define UA_TPB 256
#define UA_MAX_GT 128

#if __has_builtin(__builtin_amdgcn_global_load_async_to_lds_b128)
#define UA_HAVE_ASYNC 1
#else
#define UA_HAVE_ASYNC 0
#endif

// b128 async copy moves 16 bytes/lane; builtin wants int-vector-16B pointees.
typedef int ua_v4i __attribute__((vector_size(16)));
typedef __attribute__((address_space(1))) ua_v4i* ua_gptr_v4i;
typedef __attribute__((address_space(3))) ua_v4i* ua_lptr_v4i;

__device__ __forceinline__ void ua_wait_async() {
#if UA_HAVE_ASYNC
#if __has_builtin(__builtin_amdgcn_s_wait_asynccnt)
  __builtin_amdgcn_s_wait_asynccnt(0);
#else
  asm volatile("s_wait_asynccnt 0" ::: "memory");
#endif
#endif
}

__device__ __forceinline__ float ua_iou(const float4 a, const float4 b) {
  float lx = fmaxf(a.x, b.x);
  float ly = fmaxf(a.y, b.y);
  float rx = fminf(a.z, b.z);
  float ry = fminf(a.w, b.w);
  float iw = fmaxf(rx - lx, 0.0f);
  float ih = fmaxf(ry - ly, 0.0f);
  float inter = iw * ih;
  float aa = (a.z - a.x) * (a.w - a.y);
  float ab = (b.z - b.x) * (b.w - b.y);
  float uni = aa + ab - inter;
  return inter / fmaxf(uni, 1e-7f);
}

// ---------------------------------------------------------------------------
// Kernel 1: per-anchor max IoU over all GTs; assigned[i] = (max<0.7) ? 0 : -1
// GT table (2 KB) staged into LDS via CDNA5 async global->LDS copy.
// ---------------------------------------------------------------------------
__global__ __launch_bounds__(UA_TPB) void ua_base_init(
    const float4* __restrict__ grid, const float4* __restrict__ gt,
    int* __restrict__ assigned, int N, int M) {
  __shared__ float4 sgt[UA_MAX_GT];
  const int tid = threadIdx.x;
  if (tid < M) {
#if UA_HAVE_ASYNC
    __builtin_amdgcn_global_load_async_to_lds_b128(
        (ua_gptr_v4i)(gt + tid),
        (ua_lptr_v4i)(sgt + tid),
        /*offset=*/0, /*cpol=*/0);
#else
    sgt[tid] = gt[tid];
#endif
  }
  ua_wait_async();
  __syncthreads();

  const int i = blockIdx.x * blockDim.x + tid;
  if (i >= N) return;
  const float4 a = grid[i];
  float mx = 0.0f;  // IoU >= 0 always
  for (int m = 0; m < M; ++m) mx = fmaxf(mx, ua_iou(a, sgt[m]));
  assigned[i] = (mx < UA_NEG_THR) ? 0 : -1;
}

// ---------------------------------------------------------------------------
// Kernel 2: per-GT top-4 anchors by IoU (tie-break: smaller anchor index),
// then scatter-max of (m+1) for entries with IoU >= 0.15.
// One block per GT; private sorted top-4 per thread, LDS merge tree.
// ---------------------------------------------------------------------------
__global__ __launch_bounds__(UA_TPB_TOPK) void ua_topk_scatter(
    const float4* __restrict__ grid, const float4* __restrict__ gt,
    int* __restrict__ assigned, int N) {
  __shared__ float sv[UA_TPB_TOPK * 4];
  __shared__ int   si[UA_TPB_TOPK * 4];

  const int m   = blockIdx.x;
  const int tid = threadIdx.x;
  const float4 g = gt[m];

  // Private descending top-4 (static indices -> VGPRs). Ascending scan order
  // + strict '>' inserts give smaller-index-first on exact value ties.
  float tv0 = -1.0f, tv1 = -1.0f, tv2 = -1.0f, tv3 = -1.0f;
  int   ti0 = INT_MAX, ti1 = INT_MAX, ti2 = INT_MAX, ti3 = INT_MAX;

  for (int i = tid; i < N; i += UA_TPB_TOPK) {
    const float v = ua_iou(grid[i], g);
    if (v > tv3) {
      if (v > tv0) {
        tv3 = tv2; ti3 = ti2; tv2 = tv1; ti2 = ti1; tv1 = tv0; ti1 = ti0;
        tv0 = v;   ti0 = i;
      } else if (v > tv1) {
        tv3 = tv2; ti3 = ti2; tv2 = tv1; ti2 = ti1;
        tv1 = v;   ti1 = i;
      } else if (v > tv2) {
        tv3 = tv2; ti3 = ti2;
        tv2 = v;   ti2 = i;
      } else {
        tv3 = v;   ti3 = i;
      }
    }
  }

  const int base = tid * 4;
  sv[base + 0] = tv0; si[base + 0] = ti0;
  sv[base + 1] = tv1; si[base + 1] = ti1;
  sv[base + 2] = tv2; si[base + 2] = ti2;
  sv[base + 3] = tv3; si[base + 3] = ti3;
  __syncthreads();

  // Tree merge of sorted 4-lists: larger value wins; on ties, smaller index.
  for (int s = UA_TPB_TOPK / 2; s > 0; s >>= 1) {
    if (tid < s) {
      const int a0 = tid * 4;
      const int b0 = (tid + s) * 4;
      float ov[4]; int oi[4];
      int pa = 0, pb = 0;
      #pragma unroll
      for (int k = 0; k < 4; ++k) {
        const int ja = a0 + pa;                    // pa <= k <= 3, in range
        const int jb = b0 + ((pb > 3) ? 3 : pb);   // clamp to stay in LDS range
        const float va = sv[ja], vb = sv[jb];
        const int   ia = si[ja], ib = si[jb];
        const bool takeA =
            (pb >= 4) || (va > vb) || (va == vb && ia < ib);
        if (takeA) { ov[k] = va; oi[k] = ia; ++pa; }
        else       { ov[k] = vb; oi[k] = ib; ++pb; }
      }
      #pragma unroll
      for (int k = 0; k < 4; ++k) { sv[a0 + k] = ov[k]; si[a0 + k] = oi[k]; }
    }
    __syncthreads();
  }

  if (tid < 4) {
    const float v = sv[tid];
    const int   idx = si[tid];
    if (v >= UA_POS_THR && idx >= 0 && idx < N) {
      atomicMax(&assigned[idx], m + 1);
    }
  }
}

// ---------------------------------------------------------------------------
// Kernel 3: gather labels / bboxes.
//   a > 0  -> label = gt_labels[a-1], bbox = gt_bboxes[a-1]
//   a == 0 -> label = 0,  bbox = -1
//   a < 0  -> label = -1, bbox = -1
// ---------------------------------------------------------------------------
__global__ __launch_bounds__(UA_TPB) void ua_finalize(
    const int* __restrict__ assigned, const float4* __restrict__ gt,
    const int* __restrict__ labels, float* __restrict__ out, int N) {
  const int i = blockIdx.x * blockDim.x + threadIdx.x;
  if (i >= N) return;
  const int a = assigned[i];
  float lab;
  float4 bb;
  if (a > 0) {
    lab = (float)labels[a - 1];
    bb = gt[a - 1];
  } else {
    lab = (a == 0) ? 0.0f : -1.0f;
    bb = make_float4(-1.0f, -1.0f, -1.0f, -1.0f);
  }
  out[i] = lab;
  ((float4*)(out + N))[i] = bb;   // N*4 bytes offset is 16B aligned for N=200000
}

// ---------------------------------------------------------------------------
extern "C" void kernel_launch(void* const* d_in, const int* in_sizes, int n_in,
                              void* d_out, int out_size, void* d_ws, size_t ws_size,
                              hipStream_t stream) {
  const float4* grid   = (const float4*)d_in[0];
  const float4* gt     = (const float4*)d_in[1];
  const int*    labels = (const int*)d_in[2];
  const int N = in_sizes[0] / 4;
  const int M = in_sizes[1] / 4;   // == 128
  float* out = (float*)d_out;
  int* assigned = (int*)d_ws;      // N ints of scratch

  const int nblk = (N + UA_TPB - 1) / UA_TPB;
  ua_base_init<<<nblk, UA_TPB, 0, stream>>>(grid, gt, assigned, N, M);
  ua_topk_scatter<<<M, UA_TPB_TOPK, 0, stream>>>(grid, gt, assigned, N);
  ua_finalize<<<nblk, UA_TPB, 0, stream>>>(assigned, gt, labels, out, N);
}